// KoleoLoss_25477746000271
// MI455X (gfx1250) — compile-verified
//
#include <hip/hip_runtime.h>
#include <math.h>

// KoleoLoss for x: [8192, 512] fp32.
// Fused: row-normalize -> Gram via V_WMMA_F32_16X16X4_F32 -> per-row min
// distance (never materializing the 8192x8192 matrix) -> -mean(log(min+eps)).

typedef float v2f __attribute__((ext_vector_type(2)));
typedef float v8f __attribute__((ext_vector_type(8)));

#define N_ROWS 8192
#define N_COLS 512
#define BT 128              // block tile (128x128 Gram tile per workgroup)
#define KB 16               // k panel staged in LDS per iteration
#define LDS_STRIDE (KB + 2) // even (8B-aligned v2f reads) and conflict-free

// ---------------------------------------------------------------------------
// Pass 1: L2-normalize rows, emit xn, sq_norms(xn), init per-row min to +MAX
// ---------------------------------------------------------------------------
__global__ __launch_bounds__(256) void koleo_normalize(
    const float* __restrict__ x, float* __restrict__ xn,
    float* __restrict__ sqn, unsigned int* __restrict__ minbits) {
  __shared__ float partial[8];
  __shared__ float s_scale, s_sqn;

  const int row = blockIdx.x;
  const int tid = threadIdx.x;
  const float* xr = x + (size_t)row * N_COLS;

  float a = xr[tid];
  float b = xr[tid + 256];
  float p = a * a + b * b;
#pragma unroll
  for (int off = 16; off > 0; off >>= 1) p += __shfl_xor(p, off, 32);
  if ((tid & 31) == 0) partial[tid >> 5] = p;
  __syncthreads();
  if (tid == 0) {
    float s = 0.f;
#pragma unroll
    for (int i = 0; i < 8; ++i) s += partial[i];
    float nrm = sqrtf(s);
    float scale = 1.0f / fmaxf(nrm, 1e-12f);  // torch F.normalize clamp
    s_scale = scale;
    s_sqn = s * scale * scale;                // sum(xn*xn)
  }
  __syncthreads();
  const float scale = s_scale;
  float* xnr = xn + (size_t)row * N_COLS;
  xnr[tid] = a * scale;
  xnr[tid + 256] = b * scale;
  if (tid == 0) {
    sqn[row] = s_sqn;
    minbits[row] = 0x7f7fffffu;  // FLT_MAX bit pattern (non-neg float order)
  }
}

// ---------------------------------------------------------------------------
// Pass 2: 128x128 Gram tile per block via fp32 WMMA; fused distance + row-min
// ---------------------------------------------------------------------------
__global__ __launch_bounds__(256) void koleo_gram_min(
    const float* __restrict__ xn, const float* __restrict__ sqn,
    unsigned int* __restrict__ minbits) {
  __shared__ __align__(16) float Alds[BT * LDS_STRIDE];
  __shared__ __align__(16) float Blds[BT * LDS_STRIDE];

  const int tid  = threadIdx.x;
  const int w    = tid >> 5;        // wave id 0..7 -> 16-row strip of tile
  const int lane = tid & 31;
  const int half = lane >> 4;       // 0: lanes 0-15, 1: lanes 16-31
  const int l16  = lane & 15;

  const int i_base = blockIdx.y * BT;
  const int j_base = blockIdx.x * BT;

  v8f acc[8];
#pragma unroll
  for (int t = 0; t < 8; ++t)
#pragma unroll
    for (int e = 0; e < 8; ++e) acc[t][e] = 0.0f;

  // cooperative load mapping: thread -> (row 0..127, 8-float half of KB=16)
  const int ldr = tid >> 1;
  const int ldc = (tid & 1) * 8;

  for (int ko = 0; ko < N_COLS; ko += KB) {
    __syncthreads();  // previous compute done before LDS overwrite
    const float* ga = xn + (size_t)(i_base + ldr) * N_COLS + ko + ldc;
    const float* gb = xn + (size_t)(j_base + ldr) * N_COLS + ko + ldc;
    float4 a0 = *(const float4*)(ga);
    float4 a1 = *(const float4*)(ga + 4);
    float4 b0 = *(const float4*)(gb);
    float4 b1 = *(const float4*)(gb + 4);
    float2* la = (float2*)&Alds[ldr * LDS_STRIDE + ldc];
    float2* lb = (float2*)&Blds[ldr * LDS_STRIDE + ldc];
    la[0] = make_float2(a0.x, a0.y);
    la[1] = make_float2(a0.z, a0.w);
    la[2] = make_float2(a1.x, a1.y);
    la[3] = make_float2(a1.z, a1.w);
    lb[0] = make_float2(b0.x, b0.y);
    lb[1] = make_float2(b0.z, b0.w);
    lb[2] = make_float2(b1.x, b1.y);
    lb[3] = make_float2(b1.z, b1.w);
    __syncthreads();

#pragma unroll
    for (int kc = 0; kc < KB; kc += 4) {
      // f32 A 16x4 layout: lane l, vgpr v -> [M = l%16][K = 2*(l/16)+v]
      // => contiguous float2 per lane.  B 4x16 mirrors with N = l%16.
      const int koff = kc + 2 * half;
      v2f afrag = *(const v2f*)&Alds[(w * 16 + l16) * LDS_STRIDE + koff];
#pragma unroll
      for (int t = 0; t < 8; ++t) {
        v2f bfrag = *(const v2f*)&Blds[(t * 16 + l16) * LDS_STRIDE + koff];
        acc[t] = __builtin_amdgcn_wmma_f32_16x16x4_f32(
            /*neg_a=*/false, afrag, /*neg_b=*/false, bfrag,
            /*c_mod=*/(short)0, acc[t], /*reuse_a=*/false, /*reuse_b=*/false);
      }
    }
  }

  // Epilogue: distance + row-min.  C/D layout: vgpr v, lane l ->
  // [M = v + 8*(l/16)][N = l%16] within 16x16 tile t (cols j_base+16t..).
  float sqi[8], sqj[8];
#pragma unroll
  for (int v = 0; v < 8; ++v) sqi[v] = sqn[i_base + w * 16 + v + 8 * half];
#pragma unroll
  for (int t = 0; t < 8; ++t) sqj[t] = sqn[j_base + t * 16 + l16];

#pragma unroll
  for (int v = 0; v < 8; ++v) {
    const int gi = i_base + w * 16 + v + 8 * half;
    float dv = 3.0e38f;
#pragma unroll
    for (int t = 0; t < 8; ++t) {
      const int gj = j_base + t * 16 + l16;
      float sq = sqi[v] + sqj[t] - 2.0f * acc[t][v];
      float d = sqrtf(fmaxf(sq, 0.0f));        // safe sqrt (sqrt(0)=0)
      if (gi == gj) d = 3.0e38f;               // exclude diagonal
      dv = fminf(dv, d);
    }
    // min across the 16-lane row group (stays within each half-wave)
#pragma unroll
    for (int off = 1; off < 16; off <<= 1) dv = fminf(dv, __shfl_xor(dv, off, 32));
    if (l16 == 0)
      atomicMin(&minbits[gi], __float_as_uint(dv));  // dv >= 0: uint order ok
  }
}

// ---------------------------------------------------------------------------
// Pass 3: loss = -mean(log(min_dist + 1e-8))
// ---------------------------------------------------------------------------
__global__ __launch_bounds__(256) void koleo_finalize(
    const unsigned int* __restrict__ minbits, float* __restrict__ out) {
  __shared__ float part[8];
  const int tid = threadIdx.x;
  float s = 0.f;
  for (int r = tid; r < N_ROWS; r += 256)
    s += logf(__uint_as_float(minbits[r]) + 1e-8f);
#pragma unroll
  for (int off = 16; off > 0; off >>= 1) s += __shfl_xor(s, off, 32);
  if ((tid & 31) == 0) part[tid >> 5] = s;
  __syncthreads();
  if (tid == 0) {
    float tot = 0.f;
#pragma unroll
    for (int i = 0; i < 8; ++i) tot += part[i];
    out[0] = -tot / (float)N_ROWS;
  }
}

extern "C" void kernel_launch(void* const* d_in, const int* in_sizes, int n_in,
                              void* d_out, int out_size, void* d_ws, size_t ws_size,
                              hipStream_t stream) {
  (void)in_sizes; (void)n_in; (void)out_size; (void)ws_size;
  const float* x = (const float*)d_in[0];
  float* out = (float*)d_out;

  // workspace layout: xn [8192*512] | sqn [8192] | minbits [8192]  (~16.1 MB)
  float* xn = (float*)d_ws;
  float* sqn = xn + (size_t)N_ROWS * N_COLS;
  unsigned int* minbits = (unsigned int*)(sqn + N_ROWS);

  koleo_normalize<<<N_ROWS, 256, 0, stream>>>(x, xn, sqn, minbits);
  dim3 grid(N_ROWS / BT, N_ROWS / BT);
  koleo_gram_min<<<grid, 256, 0, stream>>>(xn, sqn, minbits);
  koleo_finalize<<<1, 256, 0, stream>>>(minbits, out);
}